// AveragePooling_21131239097224
// MI455X (gfx1250) — compile-verified
//
#include <hip/hip_runtime.h>
#include <hip/hip_bf16.h>

typedef __attribute__((ext_vector_type(2))) float v2f;
typedef __attribute__((ext_vector_type(8))) float v8f;

#define H_DIM 256
#define EPSV 1e-8f

// ---------------------------------------------------------------------------
// Kernel 1: zero the workspace (sums [S*H] + counts [S]) — must run every call.
// ---------------------------------------------------------------------------
__global__ void zero_ws_kernel(float* __restrict__ p, int n) {
    int i = blockIdx.x * blockDim.x + threadIdx.x;
    if (i < n) p[i] = 0.0f;
}

// ---------------------------------------------------------------------------
// Kernel 2: scatter-add. One block (256 threads) per input row; coalesced
// 1 KB row read, 256 float atomics into L2-resident sums, 1 count atomic.
// ---------------------------------------------------------------------------
__global__ __launch_bounds__(H_DIM)
void scatter_kernel(const float* __restrict__ x,
                    const int*   __restrict__ dst_idx,
                    float* __restrict__ sums,
                    float* __restrict__ counts) {
    const int row = blockIdx.x;
    const int tid = threadIdx.x;
    const int seg = dst_idx[row];
    const float v = x[(size_t)row * H_DIM + tid];
    atomicAdd(&sums[(size_t)seg * H_DIM + tid], v);
    if (tid == 0) atomicAdd(&counts[seg], 1.0f);
}

// ---------------------------------------------------------------------------
// Kernel 3: pooled = sums/(cnt+eps), then out = pooled @ W^T + b using
// V_WMMA_F32_16X16X4_F32. One wave per 16x16 output tile; K-loop of 64.
//
// Lane decomposition (wave32): m = lane&15, h = lane>>4.
//   A (16x4 f32): a.x = A[m][2h], a.y = A[m][2h+1]          (ISA 7.12.2)
//   B (4x16 f32): b.x = B[2h][n], b.y = B[2h+1][n], n=lane&15
//   C/D (16x16):  c[r] = D[r + 8h][m-as-column]
// ---------------------------------------------------------------------------
__global__ __launch_bounds__(256)
void pool_gemm_kernel(const float* __restrict__ sums,
                      const float* __restrict__ counts,
                      const float* __restrict__ W,
                      const float* __restrict__ bias,
                      float* __restrict__ out,
                      int S) {
    const int wid  = (int)((blockIdx.x * blockDim.x + threadIdx.x) >> 5);
    const int lane = (int)(threadIdx.x & 31);
    const int tilesN = H_DIM / 16;           // 16 tiles along output features
    const int tileM = wid / tilesN;          // segment-tile index
    const int tileN = wid % tilesN;          // feature-tile index
    if (tileM * 16 >= S) return;             // wave-uniform: EXEC stays all-1s

    const int m = lane & 15;                 // A row / B,C,D column within tile
    const int h = lane >> 4;                 // K-half selector

    const int seg = tileM * 16 + m;          // this lane's pooled row (for A)
    const int col = tileN * 16 + m;          // this lane's output column (for B, bias)

    const float inv = 1.0f / (counts[seg] + EPSV);

    const float* __restrict__ srow = sums + (size_t)seg * H_DIM; // pooled row (A)
    const float* __restrict__ wcol = W    + (size_t)col * H_DIM; // W row j == B column j

    v8f c;
    const float bj = bias[col];
#pragma unroll
    for (int r = 0; r < 8; ++r) c[r] = bj;

#pragma unroll 8
    for (int k = 0; k < H_DIM; k += 4) {
        v2f a = *(const v2f*)(srow + k + 2 * h);   // A[m][k+2h], A[m][k+2h+1]
        a.x *= inv;
        a.y *= inv;
        const v2f bf = *(const v2f*)(wcol + k + 2 * h); // B[k+2h][col], B[k+2h+1][col]
        c = __builtin_amdgcn_wmma_f32_16x16x4_f32(
                /*neg_a=*/false, a, /*neg_b=*/false, bf,
                /*c_mod=*/(short)0, c, /*reuse_a=*/false, /*reuse_b=*/false);
    }

#pragma unroll
    for (int r = 0; r < 8; ++r) {
        out[(size_t)(tileM * 16 + r + 8 * h) * H_DIM + tileN * 16 + m] = c[r];
    }
}

// ---------------------------------------------------------------------------
// Host launcher
// ---------------------------------------------------------------------------
extern "C" void kernel_launch(void* const* d_in, const int* in_sizes, int n_in,
                              void* d_out, int out_size, void* d_ws, size_t ws_size,
                              hipStream_t stream) {
    const float* x       = (const float*)d_in[0];
    const int*   dst_idx = (const int*)  d_in[1];
    // d_in[2] = dst_size (scalar) — dimensions derived host-side instead
    const float* W       = (const float*)d_in[3];
    const float* bias    = (const float*)d_in[4];
    float*       out     = (float*)d_out;

    const int N = in_sizes[0] / H_DIM;       // 100000
    const int S = out_size   / H_DIM;        // 4096

    float* sums   = (float*)d_ws;            // S*H floats
    float* counts = sums + (size_t)S * H_DIM; // S floats

    // 1) zero sums + counts
    const int zn = S * H_DIM + S;
    zero_ws_kernel<<<(zn + 255) / 256, 256, 0, stream>>>(sums, zn);

    // 2) scatter-add (dominant, HBM-bound phase: ~102 MB streamed)
    scatter_kernel<<<N, H_DIM, 0, stream>>>(x, dst_idx, sums, counts);

    // 3) fused mean + fp32 WMMA GEMM: one wave per 16x16 output tile
    const int tiles = (S / 16) * (H_DIM / 16);          // 4096 waves
    pool_gemm_kernel<<<tiles / 8, 256, 0, stream>>>(sums, counts, W, bias, out, S);
}